// AdaptiveGCN_21165598834779
// MI455X (gfx1250) — compile-verified
//
#include <hip/hip_runtime.h>
#include <hip/hip_bf16.h>

typedef __attribute__((ext_vector_type(2))) float v2f;
typedef __attribute__((ext_vector_type(8))) float v8f;

#define BDIM 16
#define CDIM 64
#define KDIM 207
#define LDIM 64
#define KP   208                    // K padded to 13*16
#define KL   (KDIM * LDIM)          // 13248 (x / y channel stride)
#define KLP  (KP * LDIM)            // 13312 (padded ws channel stride)
#define TSP  ((size_t)BDIM * CDIM * KLP) // elements per padded intermediate tensor
#define ASTR 210                    // LDS row stride for A (208x208 zero-padded)
#define XTS  212                    // LDS col stride for transposed X (64 x 208)
#define WSTR 334                    // LDS row stride for W (64x320), even + conflict-free

// D = A(16x4) * B(4x16) + C, fp32 WMMA
static __device__ __forceinline__ v8f wmma4(v2f a, v2f b, v8f c) {
    return __builtin_amdgcn_wmma_f32_16x16x4_f32(false, a, false, b, (short)0, c,
                                                 false, false);
}

// ---------------------------------------------------------------------------
// Kernel 1: per (b, adj, c) compute X1 = A @ X and X2 = A @ X1 (A: 207x207,
// X: 207x64). A is staged zero-padded in LDS (reused by both GEMMs); X and X1
// are kept TRANSPOSED in LDS so each WMMA B-operand pair is one aligned
// ds_load_b64. Outputs go to ws as (g, b, c, k[208], l), g = adj*2 + order,
// with 208-row padding so the epilogue stores need no bounds guards.
// Uses ~277 KB of the WGP's 320 KB LDS.
// ---------------------------------------------------------------------------
__global__ __launch_bounds__(256) void AdaptiveGCN_diffuse_kernel(
    const float* __restrict__ x, const float* __restrict__ a0,
    const float* __restrict__ a1, float* __restrict__ ws) {
  __shared__ float Asm[KP * ASTR];    // 174,720 B
  __shared__ float XT[LDIM * XTS];    //  54,272 B  XT[col*XTS + k]
  __shared__ float X1T[LDIM * XTS];   //  54,272 B

  const int tid = threadIdx.x;
  const int c   = blockIdx.x;
  const int adj = blockIdx.y;
  const int b   = blockIdx.z;

  const float* __restrict__ A = (adj == 0 ? a0 : a1) + (size_t)b * KDIM * KDIM;
  const float* __restrict__ xg = x + ((size_t)(b * CDIM + c)) * KL;
  float* __restrict__ w1 = ws + (size_t)(adj * 2 + 0) * TSP + ((size_t)(b * CDIM + c)) * KLP;
  float* __restrict__ w2 = ws + (size_t)(adj * 2 + 1) * TSP + ((size_t)(b * CDIM + c)) * KLP;

  // Stage A zero-padded (208x208 logical in stride-210 rows).
  for (int idx = tid; idx < KP * KP; idx += 256) {
    const int row = idx / KP, col = idx - row * KP;
    const bool v  = (row < KDIM) & (col < KDIM);
    const int  gi = v ? (row * KDIM + col) : 0;       // safe address, no exec branch
    const float t = A[gi];
    Asm[row * ASTR + col] = v ? t : 0.0f;
  }
  // Stage X transposed (kl index == row*64+col); rows >= 207 zero.
  for (int idx = tid; idx < KP * LDIM; idx += 256) {
    const int row = idx >> 6, col = idx & 63;
    const bool v  = row < KDIM;
    const float t = xg[v ? idx : 0];
    XT[col * XTS + row] = v ? t : 0.0f;
  }
  __syncthreads();

  const int wave = tid >> 5, lane = tid & 31;
  const int m = lane & 15, h = lane >> 4;

  // ---------------- GEMM1: X1 = A @ X ----------------
  for (int jt = wave; jt < 13; jt += 8) {
    v8f acc0 = {}, acc1 = {}, acc2 = {}, acc3 = {};
    const float* __restrict__ arow = &Asm[(jt * 16 + m) * ASTR];
#pragma unroll 2
    for (int ks = 0; ks < 52; ++ks) {
      const int k = ks * 4 + 2 * h;                   // even -> 8B aligned pairs
      v2f a;
      a.x = arow[k];
      a.y = arow[k + 1];
      v2f b0 = *(const v2f*)&XT[(m)      * XTS + k];
      v2f b1 = *(const v2f*)&XT[(16 + m) * XTS + k];
      v2f b2 = *(const v2f*)&XT[(32 + m) * XTS + k];
      v2f b3 = *(const v2f*)&XT[(48 + m) * XTS + k];
      acc0 = wmma4(a, b0, acc0);
      acc1 = wmma4(a, b1, acc1);
      acc2 = wmma4(a, b2, acc2);
      acc3 = wmma4(a, b3, acc3);
    }
    // Write transposed to LDS (per-lane contiguous v2f stores) + global ws.
    const int jrow = jt * 16 + 8 * h;
#pragma unroll
    for (int r = 0; r < 8; r += 2) {
      v2f t0, t1, t2, t3;
      t0.x = acc0[r]; t0.y = acc0[r + 1];
      t1.x = acc1[r]; t1.y = acc1[r + 1];
      t2.x = acc2[r]; t2.y = acc2[r + 1];
      t3.x = acc3[r]; t3.y = acc3[r + 1];
      *(v2f*)&X1T[(m)      * XTS + jrow + r] = t0;
      *(v2f*)&X1T[(16 + m) * XTS + jrow + r] = t1;
      *(v2f*)&X1T[(32 + m) * XTS + jrow + r] = t2;
      *(v2f*)&X1T[(48 + m) * XTS + jrow + r] = t3;
    }
#pragma unroll
    for (int r = 0; r < 8; ++r) {
      const int row = jrow + r;        // rows >= 207 are exact zeros, stored to pad
      w1[row * 64 + m]      = acc0[r];
      w1[row * 64 + 16 + m] = acc1[r];
      w1[row * 64 + 32 + m] = acc2[r];
      w1[row * 64 + 48 + m] = acc3[r];
    }
  }
  __syncthreads();

  // ---------------- GEMM2: X2 = A @ X1 ----------------
  for (int jt = wave; jt < 13; jt += 8) {
    v8f acc0 = {}, acc1 = {}, acc2 = {}, acc3 = {};
    const float* __restrict__ arow = &Asm[(jt * 16 + m) * ASTR];
#pragma unroll 2
    for (int ks = 0; ks < 52; ++ks) {
      const int k = ks * 4 + 2 * h;
      v2f a;
      a.x = arow[k];
      a.y = arow[k + 1];
      v2f b0 = *(const v2f*)&X1T[(m)      * XTS + k];
      v2f b1 = *(const v2f*)&X1T[(16 + m) * XTS + k];
      v2f b2 = *(const v2f*)&X1T[(32 + m) * XTS + k];
      v2f b3 = *(const v2f*)&X1T[(48 + m) * XTS + k];
      acc0 = wmma4(a, b0, acc0);
      acc1 = wmma4(a, b1, acc1);
      acc2 = wmma4(a, b2, acc2);
      acc3 = wmma4(a, b3, acc3);
    }
    const int jrow = jt * 16 + 8 * h;
#pragma unroll
    for (int r = 0; r < 8; ++r) {
      const int row = jrow + r;
      w2[row * 64 + m]      = acc0[r];
      w2[row * 64 + 16 + m] = acc1[r];
      w2[row * 64 + 32 + m] = acc2[r];
      w2[row * 64 + 48 + m] = acc3[r];
    }
  }
}

// ---------------------------------------------------------------------------
// Kernel 2: 1x1 conv. Per batch: Y(64 x 13248) = W(64 x 320) @ H(320 x 13248)
// where H channels are [x (stride 13248), ws_g0..g3 (stride 13312)]. Each wave
// computes a 64x16 column strip (4 row tiles x 80 k-steps of 16x16x4 WMMA),
// adds bias. W fragments are aligned ds_load_b64; next channel prefetched.
// ---------------------------------------------------------------------------
__global__ __launch_bounds__(256) void AdaptiveGCN_conv_kernel(
    const float* __restrict__ x, const float* __restrict__ W,
    const float* __restrict__ bias, const float* __restrict__ ws,
    float* __restrict__ y) {
  __shared__ float Wsh[CDIM * WSTR];   // 85,504 B

  const int tid   = threadIdx.x;
  const int batch = blockIdx.y;

  for (int idx = tid; idx < 64 * 320; idx += 256) {
    const int o = idx / 320, i = idx - o * 320;
    Wsh[o * WSTR + i] = W[idx];
  }
  __syncthreads();

  const int wave = tid >> 5, lane = tid & 31;
  const int m = lane & 15, h = lane >> 4;
  const int nt = blockIdx.x * 8 + wave;     // 828 column tiles of 16
  if (nt >= 828) return;
  const int col = nt * 16 + m;

  v8f acc0 = {}, acc1 = {}, acc2 = {}, acc3 = {};
#pragma unroll 2
  for (int ks = 0; ks < 80; ++ks) {
    const int i = ks * 4 + 2 * h;           // even; i,i+1 in one group of 64
    const int g = i >> 6;
    const size_t chs = (g == 0) ? (size_t)KL : (size_t)KLP;
    const float* __restrict__ src =
        (g == 0 ? x : ws + (size_t)(g - 1) * TSP) +
        ((size_t)(batch * CDIM + (i & 63))) * chs + col;
    __builtin_prefetch(src + 4 * chs, 0, 0);          // next k-step's channel
    v2f bb;
    bb.x = src[0];
    bb.y = src[chs];                         // channel i+1
    v2f wa0 = *(const v2f*)&Wsh[(m)      * WSTR + i];
    v2f wa1 = *(const v2f*)&Wsh[(16 + m) * WSTR + i];
    v2f wa2 = *(const v2f*)&Wsh[(32 + m) * WSTR + i];
    v2f wa3 = *(const v2f*)&Wsh[(48 + m) * WSTR + i];
    acc0 = wmma4(wa0, bb, acc0);
    acc1 = wmma4(wa1, bb, acc1);
    acc2 = wmma4(wa2, bb, acc2);
    acc3 = wmma4(wa3, bb, acc3);
  }
#pragma unroll
  for (int r = 0; r < 8; ++r) {
    const int o = r + 8 * h;
    y[((size_t)(batch * CDIM + o)) * KL + col]      = acc0[r] + bias[o];
    y[((size_t)(batch * CDIM + 16 + o)) * KL + col] = acc1[r] + bias[16 + o];
    y[((size_t)(batch * CDIM + 32 + o)) * KL + col] = acc2[r] + bias[32 + o];
    y[((size_t)(batch * CDIM + 48 + o)) * KL + col] = acc3[r] + bias[48 + o];
  }
}

extern "C" void kernel_launch(void* const* d_in, const int* in_sizes, int n_in,
                              void* d_out, int out_size, void* d_ws, size_t ws_size,
                              hipStream_t stream) {
  const float* x    = (const float*)d_in[0];
  const float* a0   = (const float*)d_in[1];
  const float* a1   = (const float*)d_in[2];
  const float* W    = (const float*)d_in[3];
  const float* bias = (const float*)d_in[4];
  // d_in[5]/d_in[6] are K, L scalars; compile-time constants here.
  float* ws = (float*)d_ws;   // needs 4 * TSP * 4 bytes ≈ 218 MB
  float* y  = (float*)d_out;

  dim3 g1(CDIM, 2, BDIM);     // (c, adjacency, batch)
  AdaptiveGCN_diffuse_kernel<<<g1, 256, 0, stream>>>(x, a0, a1, ws);

  dim3 g2(104, BDIM, 1);      // ceil(828/8) column-tile groups x batch
  AdaptiveGCN_conv_kernel<<<g2, 256, 0, stream>>>(x, W, bias, ws, y);
}